// PCRA_71957882077628
// MI455X (gfx1250) — compile-verified
//
#include <hip/hip_runtime.h>
#include <math.h>

#define BB    4
#define CIN   64
#define CH    32
#define HH    64
#define WW    64
#define NN    (HH*WW)   /* 4096 */
#define MCH   4
#define NDEPTH 3
#define KTOP  8

typedef float v2f __attribute__((ext_vector_type(2)));
typedef float v8f __attribute__((ext_vector_type(8)));

// ---------------------------------------------------------------------------
// K1: positional embedding + 1x1 Q/K projections.
// query[b][n][m] (row-major, m fastest) ; key[b][m][n] (column access coalesced)
// ---------------------------------------------------------------------------
__global__ void qk_project(const float* __restrict__ f_q, const float* __restrict__ f_k,
                           const float* __restrict__ wq,  const float* __restrict__ wk,
                           float* __restrict__ query, float* __restrict__ key) {
  int idx = blockIdx.x * blockDim.x + threadIdx.x;
  if (idx >= BB * NN) return;
  int b = idx / NN, n = idx % NN;
  int y = n / WW, x = n % WW;
  float fy = (float)(y + 1), fx = (float)(x + 1);
  const float* fqp = f_q + (size_t)b * CIN * NN + n;
  const float* fkp = f_k + (size_t)b * CIN * NN + n;
  float qd[4] = {0.f, 0.f, 0.f, 0.f};
  float kd[4] = {0.f, 0.f, 0.f, 0.f};
  const float LOG1E4 = 9.210340371976184f;  // ln(10000)
  #pragma unroll
  for (int i = 0; i < 16; ++i) {
    float inv_t = __expf(-LOG1E4 * (float)i * (1.0f / 16.0f));  // 10000^{-i/16}
    float sy, cy, sx, cx;
    __sincosf(fy * inv_t, &sy, &cy);
    __sincosf(fx * inv_t, &sx, &cx);
    float pc[4] = {sy, cy, sx, cx};
    int   cc[4] = {2 * i, 2 * i + 1, 32 + 2 * i, 33 + 2 * i};
    #pragma unroll
    for (int j = 0; j < 4; ++j) {
      int c = cc[j];
      float fqe = fqp[(size_t)c * NN] + pc[j];
      float fke = fkp[(size_t)c * NN] + pc[j];
      #pragma unroll
      for (int m = 0; m < 4; ++m) {
        qd[m] = fmaf(wq[m * CIN + c], fqe, qd[m]);
        kd[m] = fmaf(wk[m * CIN + c], fke, kd[m]);
      }
    }
  }
  float* qo = query + (size_t)(b * NN + n) * 4;
  #pragma unroll
  for (int m = 0; m < 4; ++m) {
    qo[m] = qd[m];
    key[(size_t)b * 4 * NN + (size_t)m * NN + n] = kd[m];
  }
}

// ---------------------------------------------------------------------------
// K2: bilinear 32->64 resize of map / m_k (half-pixel centers), value=exp(m)/e
// ---------------------------------------------------------------------------
__device__ __forceinline__ float bil32(const float* __restrict__ src, int oy, int ox) {
  float iy = (oy + 0.5f) * 0.5f - 0.5f;
  float ix = (ox + 0.5f) * 0.5f - 0.5f;
  float fy = floorf(iy), fx = floorf(ix);
  int y0 = (int)fy, x0 = (int)fx;
  float wy = iy - fy, wx = ix - fx;
  int y0c = min(max(y0, 0), 31), y1c = min(max(y0 + 1, 0), 31);
  int x0c = min(max(x0, 0), 31), x1c = min(max(x0 + 1, 0), 31);
  float v00 = src[y0c * 32 + x0c], v01 = src[y0c * 32 + x1c];
  float v10 = src[y1c * 32 + x0c], v11 = src[y1c * 32 + x1c];
  float a = v00 + wx * (v01 - v00);
  float c = v10 + wx * (v11 - v10);
  return a + wy * (c - a);
}

__global__ void resize_value(const float* __restrict__ mapi, const float* __restrict__ m_k,
                             float* __restrict__ map_r, float* __restrict__ value) {
  int idx = blockIdx.x * blockDim.x + threadIdx.x;
  if (idx >= BB * NN) return;
  int b = idx / NN, n = idx % NN;
  int y = n / WW, x = n % WW;
  map_r[idx] = bil32(mapi + (size_t)b * 1024, y, x);
  value[idx] = __expf(bil32(m_k + (size_t)b * 1024, y, x) - 1.0f);  // exp(v)/e
}

// ---------------------------------------------------------------------------
// K3: column softmax stats via V_WMMA_F32_16X16X4_F32 (online logsumexp).
// ---------------------------------------------------------------------------
__global__ void col_stats(const float* __restrict__ query, const float* __restrict__ key,
                          float* __restrict__ colmax, float* __restrict__ colsum) {
  int lane = threadIdx.x;          // blockDim = 32 (one wave)
  int tile = blockIdx.x;           // 0..255 column tile
  int b    = blockIdx.y;
  int nloc = lane & 15;
  int col  = tile * 16 + nloc;
  int khi  = (lane >> 4) << 1;     // 0 for lanes 0-15, 2 for lanes 16-31
  const float* kp = key + (size_t)b * 4 * NN;
  v2f bv;
  bv.x = kp[(size_t)khi * NN + col];
  bv.y = kp[(size_t)(khi + 1) * NN + col];
  const float* qp = query + (size_t)b * NN * 4;
  float runm = -INFINITY, runs = 0.0f;
  for (int qt = 0; qt < NN / 16; ++qt) {
    const float* ap = qp + (size_t)(qt * 16 + nloc) * 4 + khi;
    __builtin_prefetch(ap + 64, 0, 0);          // next tile -> global_prefetch_b8
    v2f av = *(const v2f*)ap;                   // 8B aligned (khi even)
    v8f cz = {0.f, 0.f, 0.f, 0.f, 0.f, 0.f, 0.f, 0.f};
    v8f c = __builtin_amdgcn_wmma_f32_16x16x4_f32(
        false, av, false, bv, (short)0, cz, false, false);
    float tm = c[0];
    #pragma unroll
    for (int i = 1; i < 8; ++i) tm = fmaxf(tm, c[i]);
    tm = fmaxf(tm, __shfl_xor(tm, 16, 32));
    float newm = fmaxf(runm, tm);
    float ps = 0.0f;
    #pragma unroll
    for (int i = 0; i < 8; ++i) ps += __expf(c[i] - newm);
    ps += __shfl_xor(ps, 16, 32);
    runs = runs * __expf(runm - newm) + ps;
    runm = newm;
  }
  if (lane < 16) {
    colmax[(size_t)b * NN + col] = runm;
    colsum[(size_t)b * NN + col] = runs;
  }
}

// ---------------------------------------------------------------------------
// K4: per-row top-8 of context = softmax_col(sim) * value. One wave per row.
// ---------------------------------------------------------------------------
__global__ void topk_rows(const float* __restrict__ query, const float* __restrict__ key,
                          const float* __restrict__ colmax, const float* __restrict__ colsum,
                          const float* __restrict__ value, float* __restrict__ ctx_m) {
  int lane = threadIdx.x & 31;
  int wid  = threadIdx.x >> 5;
  int row  = blockIdx.x * 8 + wid;
  int b = row / NN, q = row % NN;
  const float* qp = query + (size_t)(b * NN + q) * 4;
  float q0 = qp[0], q1 = qp[1], q2 = qp[2], q3 = qp[3];
  const float* kp = key    + (size_t)b * 4 * NN;
  const float* cm = colmax + (size_t)b * NN;
  const float* cs = colsum + (size_t)b * NN;
  const float* vv = value  + (size_t)b * NN;
  float t[8];
  #pragma unroll
  for (int i = 0; i < 8; ++i) t[i] = -INFINITY;
  for (int i = 0; i < NN / 32; ++i) {
    int k = (i << 5) + lane;
    float s = q0 * kp[k] + q1 * kp[NN + k] + q2 * kp[2 * NN + k] + q3 * kp[3 * NN + k];
    float ctx = __expf(s - cm[k]) * vv[k] / cs[k];
    if (ctx > t[7]) {
      t[7] = ctx;
      #pragma unroll
      for (int j = 7; j > 0; --j) {
        if (t[j] > t[j - 1]) { float tmp = t[j - 1]; t[j - 1] = t[j]; t[j] = tmp; }
      }
    }
  }
  int ptr = 0;
  float sum = 0.0f;
  #pragma unroll
  for (int r = 0; r < KTOP; ++r) {
    float cand = (ptr < 8) ? t[ptr] : -INFINITY;
    float v = cand;
    #pragma unroll
    for (int off = 16; off >= 1; off >>= 1) v = fmaxf(v, __shfl_xor(v, off, 32));
    unsigned long long msk = __ballot(cand == v);
    int winner = __ffsll(msk) - 1;
    if (lane == winner) ptr++;
    sum += v;
  }
  if (lane == 0) ctx_m[(size_t)b * NN + q] = sum * (1.0f / (float)KTOP);
}

// ---------------------------------------------------------------------------
// K5: corrected map + reverse-attention gate
// ---------------------------------------------------------------------------
__global__ void corr_rmap(const float* __restrict__ ctx_m, const float* __restrict__ map_r,
                          float* __restrict__ corrected_out, float* __restrict__ rmap) {
  int idx = blockIdx.x * blockDim.x + threadIdx.x;
  if (idx >= BB * NN) return;
  float mr = map_r[idx];
  float c = ctx_m[idx] * mr + mr;
  corrected_out[idx] = c;
  rmap[idx] = 1.0f - 1.0f / (1.0f + __expf(-c));
}

// ---------------------------------------------------------------------------
// K6: 1x1 conv fused with rmap*f  (rmap factored out of the channel sum)
// ---------------------------------------------------------------------------
__global__ void conv1x1_in(const float* __restrict__ f, const float* __restrict__ rmap,
                           const float* __restrict__ w, const float* __restrict__ bias,
                           float* __restrict__ out) {
  int idx = blockIdx.x * blockDim.x + threadIdx.x;
  if (idx >= BB * CH * NN) return;
  int b = idx / (CH * NN);
  int c = (idx / NN) % CH;
  int n = idx % NN;
  const float* fp = f + (size_t)b * CH * NN + n;
  const float* wp = w + c * CH;
  float s = 0.0f;
  #pragma unroll
  for (int ci = 0; ci < CH; ++ci) s = fmaf(wp[ci], fp[(size_t)ci * NN], s);
  out[idx] = fmaf(rmap[b * NN + n], s, bias[c]);
}

// ---------------------------------------------------------------------------
// Prep: transpose w_mid (layer,co,ci,t) -> k-major wT[layer][k=ci*9+t][co]
// ---------------------------------------------------------------------------
__global__ void transpose_wmid(const float* __restrict__ w_mid, float* __restrict__ wT) {
  int idx = blockIdx.x * blockDim.x + threadIdx.x;
  if (idx >= NDEPTH * CH * CH * 9) return;
  int l  = idx / (CH * 288);
  int rm = idx % (CH * 288);
  int co = rm / 288;
  int k  = rm % 288;            // k = ci*9 + t; w_mid flat == idx
  wT[(size_t)l * 288 * CH + (size_t)k * CH + co] = w_mid[idx];
}

// ---------------------------------------------------------------------------
// K7: 3x3 conv CH->CH as implicit GEMM on V_WMMA_F32_16X16X4_F32.
// Block = 8 waves = one image row: 4 x-strips (M=16 pixels) x 2 co-halves (N=16).
// K = 32 ci * 9 taps, padded to 12 taps/ci so each K=4 step stays in one ci:
// 96 chained WMMAs per wave. Input slab (3x66x32ch) + weights (288x32) in LDS.
// ---------------------------------------------------------------------------
__global__ void __launch_bounds__(256)
conv3x3_wmma(const float* __restrict__ in, const float* __restrict__ wT,
             const float* __restrict__ bias, float* __restrict__ out, int do_relu) {
  __shared__ float sW[288 * 32];        // 36 KB, k-major [k][co]
  __shared__ float sI[32 * 3 * 66];     // 25 KB, [ci][row 0..2][imgx+1 (0..65)]
  int tid = threadIdx.x;
  int y = blockIdx.x;                   // image row
  int b = blockIdx.y;

  // ---- stage weights into LDS via the gfx1250 async-to-LDS path ----------
#if defined(__gfx1250__) && __has_builtin(__builtin_amdgcn_global_load_async_to_lds_b32)
  typedef __attribute__((address_space(1))) int as1_int;   // global
  typedef __attribute__((address_space(3))) int as3_int;   // LDS
  for (int i = tid; i < 288 * 32; i += 256) {
    __builtin_amdgcn_global_load_async_to_lds_b32(
        (as1_int*)(wT + i), (as3_int*)(sW + i), 0, 0);
  }
  __builtin_amdgcn_s_wait_asynccnt(0);
#else
  for (int i = tid; i < 288 * 32; i += 256) sW[i] = wT[i];
#endif
  // ---- stage zero-padded input slab: rows y-1..y+1, x=-1..64, all 32 ci ----
  for (int i = tid; i < 32 * 3 * 66; i += 256) {
    int ci = i / 198;
    int rr = (i % 198) / 66;
    int xx = (i % 66) - 1;
    int yy = y + rr - 1;
    float v = 0.0f;
    if (yy >= 0 && yy < HH && xx >= 0 && xx < WW)
      v = in[(size_t)b * CH * NN + (size_t)ci * NN + yy * WW + xx];
    sI[i] = v;
  }
  __syncthreads();

  int lane = tid & 31;
  int wv   = tid >> 5;
  int x0     = (wv & 3) << 4;           // pixel strip base
  int cobase = (wv >> 2) << 4;          // out-channel half
  int nloc = lane & 15;
  int khi  = (lane >> 4) << 1;          // K split: 0/1 vs 2/3
  int co   = cobase + nloc;
  int xn   = x0 + nloc;                 // A-operand pixel column in sI coords

  // Per-wave constants for the 3 K-steps inside one ci (taps padded 9->12)
  int aoff0[3], aoff1[3], boff0[3], boff1[3];
  bool v0[3], v1[3];
  #pragma unroll
  for (int sj = 0; sj < 3; ++sj) {
    int t0 = 4 * sj + khi, t1 = t0 + 1;
    v0[sj] = (t0 < 9);
    v1[sj] = (t1 < 9);
    int dy0 = t0 / 3, dx0 = t0 - dy0 * 3;
    int dy1 = t1 / 3, dx1 = t1 - dy1 * 3;
    aoff0[sj] = v0[sj] ? (dy0 * 66 + dx0) : 0;
    aoff1[sj] = v1[sj] ? (dy1 * 66 + dx1) : 0;
    boff0[sj] = (v0[sj] ? t0 * 32 : 0) + co;
    boff1[sj] = (v1[sj] ? t1 * 32 : 0) + co;
  }

  v8f acc = {0.f, 0.f, 0.f, 0.f, 0.f, 0.f, 0.f, 0.f};
  #pragma unroll 4
  for (int ci = 0; ci < CH; ++ci) {
    int aBase = ci * 198 + xn;
    int bBase = ci * 288;
    #pragma unroll
    for (int sj = 0; sj < 3; ++sj) {
      v2f av, bv;
      av.x = v0[sj] ? sI[aBase + aoff0[sj]] : 0.0f;
      av.y = v1[sj] ? sI[aBase + aoff1[sj]] : 0.0f;
      bv.x = v0[sj] ? sW[bBase + boff0[sj]] : 0.0f;
      bv.y = v1[sj] ? sW[bBase + boff1[sj]] : 0.0f;
      acc = __builtin_amdgcn_wmma_f32_16x16x4_f32(
          false, av, false, bv, (short)0, acc, false, false);
    }
  }

  // Epilogue: D layout = M(pixel) rows over VGPRs, N(co) over lanes.
  float bsv = bias[co];
  int half = lane >> 4;                 // pixels 0..7 vs 8..15
  float4 lo, hi;
  lo.x = acc[0] + bsv; lo.y = acc[1] + bsv; lo.z = acc[2] + bsv; lo.w = acc[3] + bsv;
  hi.x = acc[4] + bsv; hi.y = acc[5] + bsv; hi.z = acc[6] + bsv; hi.w = acc[7] + bsv;
  if (do_relu) {
    lo.x = fmaxf(lo.x, 0.f); lo.y = fmaxf(lo.y, 0.f); lo.z = fmaxf(lo.z, 0.f); lo.w = fmaxf(lo.w, 0.f);
    hi.x = fmaxf(hi.x, 0.f); hi.y = fmaxf(hi.y, 0.f); hi.z = fmaxf(hi.z, 0.f); hi.w = fmaxf(hi.w, 0.f);
  }
  float* op = out + (size_t)b * CH * NN + (size_t)co * NN + y * WW + x0 + half * 8;
  *(float4*)op = lo;
  *(float4*)(op + 4) = hi;
}

// ---------------------------------------------------------------------------
// K8: final 3x3 conv CH->1 + corrected_map
// ---------------------------------------------------------------------------
__global__ void conv_out_k(const float* __restrict__ xin, const float* __restrict__ w,
                           const float* __restrict__ bias, const float* __restrict__ corrected,
                           float* __restrict__ out) {
  int idx = blockIdx.x * blockDim.x + threadIdx.x;
  if (idx >= BB * NN) return;
  int b = idx / NN, n = idx % NN;
  int y = n / WW, x = n % WW;
  const float* ip = xin + (size_t)b * CH * NN;
  float acc = bias[0];
  for (int ci = 0; ci < CH; ++ci) {
    const float* pc = ip + (size_t)ci * NN;
    const float* wc = w + ci * 9;
    #pragma unroll
    for (int dy = -1; dy <= 1; ++dy) {
      int yy = y + dy;
      if ((unsigned)yy >= (unsigned)HH) continue;
      #pragma unroll
      for (int dx = -1; dx <= 1; ++dx) {
        int xx = x + dx;
        if ((unsigned)xx >= (unsigned)WW) continue;
        acc = fmaf(wc[(dy + 1) * 3 + (dx + 1)], pc[yy * WW + xx], acc);
      }
    }
  }
  out[idx] = acc + corrected[idx];
}

// ---------------------------------------------------------------------------
extern "C" void kernel_launch(void* const* d_in, const int* in_sizes, int n_in,
                              void* d_out, int out_size, void* d_ws, size_t ws_size,
                              hipStream_t stream) {
  const float* f     = (const float*)d_in[0];
  const float* mapi  = (const float*)d_in[1];
  const float* f_q   = (const float*)d_in[2];
  const float* f_k   = (const float*)d_in[3];
  const float* m_k   = (const float*)d_in[4];
  const float* wq    = (const float*)d_in[5];
  const float* wk    = (const float*)d_in[6];
  const float* w_in  = (const float*)d_in[7];
  const float* b_in  = (const float*)d_in[8];
  const float* w_mid = (const float*)d_in[9];
  const float* b_mid = (const float*)d_in[10];
  const float* w_out = (const float*)d_in[11];
  const float* b_out = (const float*)d_in[12];
  float* out = (float*)d_out;
  float* ws  = (float*)d_ws;

  float* query  = ws;                 // 65536
  float* key    = query  + 65536;     // 65536
  float* colmax = key    + 65536;     // 16384
  float* colsum = colmax + 16384;     // 16384
  float* value  = colsum + 16384;     // 16384
  float* map_r  = value  + 16384;     // 16384
  float* ctx_m  = map_r  + 16384;     // 16384
  float* rmap   = ctx_m  + 16384;     // 16384
  float* xA     = rmap   + 16384;     // 524288
  float* xB     = xA     + 524288;    // 524288
  float* wT     = xB     + 524288;    // 27648  (total ~5.2 MB)

  float* out_x = out;                          // (4,32,64,64)
  float* out_o = out + (size_t)BB * CH * NN;   // (4,1,64,64)
  float* out_c = out_o + (size_t)BB * NN;      // (4,1,64,64)

  transpose_wmid<<<dim3((NDEPTH * CH * CH * 9 + 255) / 256), 256, 0, stream>>>(w_mid, wT);
  qk_project<<<dim3((BB * NN + 255) / 256), 256, 0, stream>>>(f_q, f_k, wq, wk, query, key);
  resize_value<<<dim3((BB * NN + 255) / 256), 256, 0, stream>>>(mapi, m_k, map_r, value);
  col_stats<<<dim3(NN / 16, BB), 32, 0, stream>>>(query, key, colmax, colsum);
  topk_rows<<<dim3(BB * NN / 8), 256, 0, stream>>>(query, key, colmax, colsum, value, ctx_m);
  corr_rmap<<<dim3((BB * NN + 255) / 256), 256, 0, stream>>>(ctx_m, map_r, out_c, rmap);
  conv1x1_in<<<dim3(BB * CH * NN / 256), 256, 0, stream>>>(f, rmap, w_in, b_in, xA);
  conv3x3_wmma<<<dim3(HH, BB), 256, 0, stream>>>(xA, wT + 0 * 288 * CH, b_mid + 0 * CH, xB, 1);
  conv3x3_wmma<<<dim3(HH, BB), 256, 0, stream>>>(xB, wT + 1 * 288 * CH, b_mid + 1 * CH, xA, 1);
  conv3x3_wmma<<<dim3(HH, BB), 256, 0, stream>>>(xA, wT + 2 * 288 * CH, b_mid + 2 * CH, out_x, 1);
  conv_out_k<<<dim3((BB * NN + 255) / 256), 256, 0, stream>>>(out_x, w_out, b_out, out_c, out_o);

  (void)in_sizes; (void)n_in; (void)out_size; (void)ws_size;
}